// GCN_24000277250640
// MI455X (gfx1250) — compile-verified
//
#include <hip/hip_runtime.h>
#include <hip/hip_bf16.h>
#include <math.h>

// ---------------- problem constants (from reference) ----------------
#define NGRAPHS   256
#define HID       64
#define NLAYERS   3
#define DCAT      192      // HID * NLAYERS
#define NCLS      10
#define GN_EPS    1e-5f

typedef __attribute__((ext_vector_type(2))) float v2f;
typedef __attribute__((ext_vector_type(8))) float v8f;

static inline int cdiv(long long a, long long b) { return (int)((a + b - 1) / b); }

__device__ __forceinline__ void atomicAddF32(float* p, float v) {
  // no-return fp32 atomic -> global_atomic_add_f32 (STOREcnt tracked)
  __hip_atomic_fetch_add(p, v, __ATOMIC_RELAXED, __HIP_MEMORY_SCOPE_AGENT);
}

// ---------------- degree / counts ----------------
__global__ void k_init_deg(float* deg, int n) {
  int i = blockIdx.x * blockDim.x + threadIdx.x;
  if (i < n) deg[i] = 1.0f;  // self-loop contributes 1
}

__global__ void k_deg_accum(const int* __restrict__ dst, float* deg, int e) {
  int i = blockIdx.x * blockDim.x + threadIdx.x;
  if (i < e) atomicAddF32(&deg[dst[i]], 1.0f);
}

__global__ void k_rsqrt_inplace(float* deg, int n) {
  int i = blockIdx.x * blockDim.x + threadIdx.x;
  if (i < n) deg[i] = rsqrtf(deg[i]);   // deg >= 1 always (self loops)
}

__global__ void k_cnt(const int* __restrict__ batch, float* cnt, int n) {
  int i = blockIdx.x * blockDim.x + threadIdx.x;
  if (i < n) atomicAddF32(&cnt[batch[i]], 1.0f);
}

__global__ void k_cnt_max1(float* cnt, int g) {
  int i = blockIdx.x * blockDim.x + threadIdx.x;
  if (i < g) cnt[i] = fmaxf(cnt[i], 1.0f);
}

// ---------------- fp32 WMMA GEMM: C[M,N] = A[M,K] @ B[K,N] (+bias)(+relu) ---
// One wave computes a 16-row strip x (TN*16) columns. Requires:
//   M % 16 == 0, N == TN*16 (compile-time), K == KDIM (compile-time, %4==0).
// USELDS: stage the whole B (KDIM x TN*16 f32) in LDS once per block and read
// fragments with ds_load in the K loop (all waves of the block share B).
template <int TN, int KDIM, bool USELDS, bool RELU>
__global__ void k_gemm_wmma_f32(const float* __restrict__ A,
                                const float* __restrict__ B,
                                const float* __restrict__ bias,
                                float* __restrict__ C,
                                int M) {
  constexpr int NDIM = TN * 16;
  __shared__ float Bs[USELDS ? (KDIM * NDIM) : 4];

  if (USELDS) {
    for (int idx = threadIdx.x * 4; idx < KDIM * NDIM; idx += blockDim.x * 4)
      *(float4*)&Bs[idx] = *(const float4*)(B + idx);
    __syncthreads();
  }

  const int lane  = threadIdx.x & 31;
  const int wave  = threadIdx.x >> 5;
  const int strip = blockIdx.x * (blockDim.x >> 5) + wave;  // 16-row strip
  if (strip * 16 >= M) return;                               // wave-uniform
  const int row0 = strip * 16;
  const int half = lane >> 4;   // 0: lanes 0-15, 1: lanes 16-31
  const int l16  = lane & 15;

  v8f acc[TN];
#pragma unroll
  for (int t = 0; t < TN; ++t)
#pragma unroll
    for (int r = 0; r < 8; ++r) acc[t][r] = 0.0f;

  const float* arow = A + (size_t)(row0 + l16) * KDIM;
  for (int k0 = 0; k0 < KDIM; k0 += 4) {
    // A fragment 16x4: VGPR0 holds K=k0+2*half, VGPR1 holds K=k0+2*half+1
    v2f a;
    a.x = arow[k0 + 2 * half];
    a.y = arow[k0 + 2 * half + 1];
#pragma unroll
    for (int t = 0; t < TN; ++t) {
      const int col = t * 16 + l16;
      v2f b;  // B fragment 4x16, mirrored layout (N striped across lanes)
      if (USELDS) {
        b.x = Bs[(k0 + 2 * half) * NDIM + col];
        b.y = Bs[(k0 + 2 * half + 1) * NDIM + col];
      } else {
        b.x = B[(size_t)(k0 + 2 * half) * NDIM + col];
        b.y = B[(size_t)(k0 + 2 * half + 1) * NDIM + col];
      }
      acc[t] = __builtin_amdgcn_wmma_f32_16x16x4_f32(
          /*neg_a=*/false, a, /*neg_b=*/false, b,
          /*c_mod=*/(short)0, acc[t], /*reuse_a=*/false, /*reuse_b=*/false);
    }
  }

#pragma unroll
  for (int t = 0; t < TN; ++t) {
    const int col = t * 16 + l16;
    float bv = bias ? bias[col] : 0.0f;
#pragma unroll
    for (int r = 0; r < 8; ++r) {
      const int row = row0 + r + 8 * half;   // C/D: VGPR r = rows r and r+8
      float v = acc[t][r] + bv;
      if (RELU) v = fmaxf(v, 0.0f);
      C[(size_t)row * NDIM + col] = v;
    }
  }
}

// ---------------- edge scatter: agg[dst] += hw[src] * dinv[src]*dinv[dst] ---
// One thread per (edge, 4 features): float4 gather + 4 no-return f32 atomics.
// 16 consecutive lanes cover one edge's 64 features -> coalesced 256B bursts.
__global__ void k_scatter(const float* __restrict__ hw,
                          const int* __restrict__ src,
                          const int* __restrict__ dst,
                          const float* __restrict__ dinv,
                          float* __restrict__ agg,
                          int E, int N) {
  long long i = (long long)blockIdx.x * blockDim.x + threadIdx.x;
  long long total = ((long long)E + N) * (HID / 4);
  if (i >= total) return;
  int q = (int)(i & 15);         // which float4 of the 64-feature row
  long long e = i >> 4;
  int s, d;
  if (e < E) { s = src[e]; d = dst[e]; }
  else       { s = d = (int)(e - E); }          // self loop
  float w = dinv[s] * dinv[d];
  const float4 v = *(const float4*)(hw + (size_t)s * HID + q * 4);
  float* p = agg + (size_t)d * HID + q * 4;
  atomicAddF32(p + 0, v.x * w);
  atomicAddF32(p + 1, v.y * w);
  atomicAddF32(p + 2, v.z * w);
  atomicAddF32(p + 3, v.w * w);
}

// ---------------- GraphNorm pieces (x4 vectorized over features) ----------
__global__ void k_seg_sum_mean(const float* __restrict__ agg,
                               const float* __restrict__ bias,
                               const int* __restrict__ batch,
                               float* __restrict__ mean, int n16) {
  int i = blockIdx.x * blockDim.x + threadIdx.x;
  if (i >= n16) return;
  int q = i & 15;
  int n = i >> 4;
  int g = batch[n];
  float4 a = *(const float4*)(agg + (size_t)n * HID + q * 4);
  float4 b = *(const float4*)(bias + q * 4);
  float* m = mean + g * HID + q * 4;
  atomicAddF32(m + 0, a.x + b.x);
  atomicAddF32(m + 1, a.y + b.y);
  atomicAddF32(m + 2, a.z + b.z);
  atomicAddF32(m + 3, a.w + b.w);
}

__global__ void k_div_by_cnt(float* buf, const float* __restrict__ cnt,
                             int rows, int F) {
  int i = blockIdx.x * blockDim.x + threadIdx.x;
  if (i < rows * F) buf[i] /= cnt[i / F];
}

__global__ void k_center_var(const float* __restrict__ agg,
                             const float* __restrict__ bias,
                             const float* __restrict__ alpha,
                             const float* __restrict__ mean,
                             const int* __restrict__ batch,
                             float* __restrict__ hc,
                             float* __restrict__ var, int n16) {
  int i = blockIdx.x * blockDim.x + threadIdx.x;
  if (i >= n16) return;
  int q = i & 15;
  int n = i >> 4;
  int g = batch[n];
  float4 a = *(const float4*)(agg + (size_t)n * HID + q * 4);
  float4 b = *(const float4*)(bias + q * 4);
  float4 al = *(const float4*)(alpha + q * 4);
  float4 mu = *(const float4*)(mean + g * HID + q * 4);
  float4 h;
  h.x = a.x + b.x - al.x * mu.x;
  h.y = a.y + b.y - al.y * mu.y;
  h.z = a.z + b.z - al.z * mu.z;
  h.w = a.w + b.w - al.w * mu.w;
  *(float4*)(hc + (size_t)n * HID + q * 4) = h;
  float* vv = var + g * HID + q * 4;
  atomicAddF32(vv + 0, h.x * h.x);
  atomicAddF32(vv + 1, h.y * h.y);
  atomicAddF32(vv + 2, h.z * h.z);
  atomicAddF32(vv + 3, h.w * h.w);
}

__global__ void k_norm_relu_pool(const float* __restrict__ hc,
                                 const float* __restrict__ var,
                                 const int* __restrict__ batch,
                                 const float* __restrict__ gamma,
                                 const float* __restrict__ beta,
                                 float* __restrict__ out,
                                 float* __restrict__ pooled,
                                 int layer, int n16) {
  int i = blockIdx.x * blockDim.x + threadIdx.x;
  if (i >= n16) return;
  int q = i & 15;
  int n = i >> 4;
  int g = batch[n];
  float4 h = *(const float4*)(hc + (size_t)n * HID + q * 4);
  float4 vv = *(const float4*)(var + g * HID + q * 4);
  float4 gm = *(const float4*)(gamma + q * 4);
  float4 bt = *(const float4*)(beta + q * 4);
  float4 o;
  o.x = fmaxf(gm.x * h.x * rsqrtf(vv.x + GN_EPS) + bt.x, 0.0f);
  o.y = fmaxf(gm.y * h.y * rsqrtf(vv.y + GN_EPS) + bt.y, 0.0f);
  o.z = fmaxf(gm.z * h.z * rsqrtf(vv.z + GN_EPS) + bt.z, 0.0f);
  o.w = fmaxf(gm.w * h.w * rsqrtf(vv.w + GN_EPS) + bt.w, 0.0f);
  *(float4*)(out + (size_t)n * HID + q * 4) = o;
  float* p = pooled + g * DCAT + layer * HID + q * 4;
  atomicAddF32(p + 0, o.x);
  atomicAddF32(p + 1, o.y);
  atomicAddF32(p + 2, o.z);
  atomicAddF32(p + 3, o.w);
}

// ---------------- classifier tail ----------------
__global__ void k_logits(const float* __restrict__ z1,
                         const float* __restrict__ w2,
                         const float* __restrict__ b2,
                         float* __restrict__ logits) {
  int i = blockIdx.x * blockDim.x + threadIdx.x;
  if (i >= NGRAPHS * NCLS) return;
  int c = i % NCLS, g = i / NCLS;
  float s = b2[c];
  const float* row = z1 + (size_t)g * DCAT;
  for (int k = 0; k < DCAT; ++k) s += row[k] * w2[k * NCLS + c];
  logits[i] = s;
}

__global__ void k_logsoftmax(float* logits) {
  int g = blockIdx.x * blockDim.x + threadIdx.x;
  if (g >= NGRAPHS) return;
  float* row = logits + g * NCLS;
  float m = row[0];
  for (int c = 1; c < NCLS; ++c) m = fmaxf(m, row[c]);
  float s = 0.0f;
  for (int c = 0; c < NCLS; ++c) s += expf(row[c] - m);
  float lse = m + logf(s);
  for (int c = 0; c < NCLS; ++c) row[c] = row[c] - lse;
}

// ---------------- launcher ----------------
extern "C" void kernel_launch(void* const* d_in, const int* in_sizes, int n_in,
                              void* d_out, int out_size, void* d_ws, size_t ws_size,
                              hipStream_t stream) {
  const float* x      = (const float*)d_in[0];
  const int*   eidx   = (const int*)d_in[1];
  const int*   batch  = (const int*)d_in[2];
  const float* conv_w = (const float*)d_in[3];
  const float* conv_b = (const float*)d_in[4];
  const float* gn_w   = (const float*)d_in[5];
  const float* gn_b   = (const float*)d_in[6];
  const float* gn_a   = (const float*)d_in[7];
  const float* w1     = (const float*)d_in[8];
  const float* b1     = (const float*)d_in[9];
  const float* w2     = (const float*)d_in[10];
  const float* b2     = (const float*)d_in[11];

  const int N = in_sizes[0] / HID;       // 100000
  const int E = in_sizes[1] / 2;         // 1200000
  const int* src = eidx;
  const int* dst = eidx + E;
  const size_t Nf = (size_t)N * HID;

  // ---- workspace carve-up (floats) ----
  float* ws     = (float*)d_ws;
  float* dinv   = ws;                    ws += N;              // deg -> rsqrt in place
  float* cnt    = ws;                    ws += NGRAPHS;
  float* mean   = ws;                    ws += NGRAPHS * HID;
  float* var    = ws;                    ws += NGRAPHS * HID;
  float* pooled = ws;                    ws += NGRAPHS * DCAT;
  float* z1     = ws;                    ws += NGRAPHS * DCAT;
  float* Y      = ws;                    ws += Nf;             // hw -> hc
  float* Z      = ws;                    ws += Nf;             // agg
  float* X      = ws;                    ws += Nf;             // layer output

  float* emb_out = (float*)d_out;                 // [N, HID]
  float* logits  = (float*)d_out + Nf;            // [G, NCLS]

  const int B256 = 256;

  // ---- degree / normalization constants ----
  k_init_deg<<<cdiv(N, B256), B256, 0, stream>>>(dinv, N);
  k_deg_accum<<<cdiv(E, B256), B256, 0, stream>>>(dst, dinv, E);
  k_rsqrt_inplace<<<cdiv(N, B256), B256, 0, stream>>>(dinv, N);

  hipMemsetAsync(cnt, 0, NGRAPHS * sizeof(float), stream);
  k_cnt<<<cdiv(N, B256), B256, 0, stream>>>(batch, cnt, N);
  k_cnt_max1<<<1, NGRAPHS, 0, stream>>>(cnt, NGRAPHS);

  hipMemsetAsync(pooled, 0, NGRAPHS * DCAT * sizeof(float), stream);

  const long long scatTot = ((long long)E + N) * (HID / 4);
  const int n16 = N * (HID / 4);

  for (int l = 0; l < NLAYERS; ++l) {
    const float* h_in = (l == 0) ? x : X;
    const float* W  = conv_w + (size_t)l * HID * HID;
    const float* cb = conv_b + l * HID;

    // hw = h_in @ W  (WMMA f32, B staged in 16KB LDS, TN=4 -> 64 cols/strip)
    {
      int strips = N / 16;               // 100000/16 = 6250 exact
      int waves  = B256 / 32;
      k_gemm_wmma_f32<4, HID, true, false>
          <<<cdiv(strips, waves), B256, 0, stream>>>(h_in, W, nullptr, Y, N);
    }

    hipMemsetAsync(Z, 0, Nf * sizeof(float), stream);
    hipMemsetAsync(mean, 0, NGRAPHS * HID * sizeof(float), stream);
    hipMemsetAsync(var, 0, NGRAPHS * HID * sizeof(float), stream);

    k_scatter<<<cdiv(scatTot, B256), B256, 0, stream>>>(Y, src, dst, dinv, Z, E, N);

    k_seg_sum_mean<<<cdiv(n16, B256), B256, 0, stream>>>(Z, cb, batch, mean, n16);
    k_div_by_cnt<<<cdiv(NGRAPHS * HID, B256), B256, 0, stream>>>(mean, cnt, NGRAPHS, HID);

    k_center_var<<<cdiv(n16, B256), B256, 0, stream>>>(
        Z, cb, gn_a + l * HID, mean, batch, Y, var, n16);
    k_div_by_cnt<<<cdiv(NGRAPHS * HID, B256), B256, 0, stream>>>(var, cnt, NGRAPHS, HID);

    float* out = (l == NLAYERS - 1) ? emb_out : X;
    k_norm_relu_pool<<<cdiv(n16, B256), B256, 0, stream>>>(
        Y, var, batch, gn_w + l * HID, gn_b + l * HID, out, pooled, l, n16);
  }

  // pooled = segment_sum(feats) / cnt
  k_div_by_cnt<<<cdiv(NGRAPHS * DCAT, B256), B256, 0, stream>>>(pooled, cnt, NGRAPHS, DCAT);

  // z1 = relu(pooled @ w1 + b1)   [256,192]@[192,192] -> TN=12, 16 strips
  {
    int blockT = 128;                    // 4 waves
    int waves  = blockT / 32;
    int strips = NGRAPHS / 16;           // 16
    k_gemm_wmma_f32<12, DCAT, false, true>
        <<<cdiv(strips, waves), blockT, 0, stream>>>(pooled, w1, b1, z1, NGRAPHS);
  }

  // logits + log_softmax (in place in d_out)
  k_logits<<<cdiv(NGRAPHS * NCLS, B256), B256, 0, stream>>>(z1, w2, b2, logits);
  k_logsoftmax<<<1, NGRAPHS, 0, stream>>>(logits);
}